// NeuroSymbolicGNN_8881992368450
// MI455X (gfx1250) — compile-verified
//
#include <hip/hip_runtime.h>
#include <hip/hip_bf16.h>

#define NN 50000
#define NE 800000
#define IN_DIM 128
#define HID_DIM 128
#define OUT_DIM 64
#define N_REL 8

typedef __attribute__((ext_vector_type(2))) float v2f;
typedef __attribute__((ext_vector_type(8))) float v8f;

// ---------------- utility kernels ----------------

__global__ void zero_u32_kernel(unsigned int* __restrict__ p, int n) {
    int i = blockIdx.x * blockDim.x + threadIdx.x;
    if (i < n) p[i] = 0u;
}

__global__ void zero_f32_kernel(float* __restrict__ p, int n) {
    int i = blockIdx.x * blockDim.x + threadIdx.x;
    if (i < n) p[i] = 0.0f;
}

__global__ void relu_f32_kernel(float* __restrict__ p, int n) {
    int i = blockIdx.x * blockDim.x + threadIdx.x;
    if (i < n) p[i] = fmaxf(p[i], 0.0f);
}

// ---------------- edge preprocessing ----------------
// counts[r*NN + dst] = #edges with relation r into dst (mean normalization)
// meta[0..7]  = per-relation edge counts
// meta[8..15] = bucket cursors
// meta[16..24]= per-relation bucket offsets (exclusive scan, meta[24] = NE)

__global__ void edge_hist_kernel(const long long* __restrict__ edge_index,
                                 const long long* __restrict__ edge_type,
                                 unsigned int* __restrict__ counts,
                                 unsigned int* __restrict__ meta) {
    int e = blockIdx.x * blockDim.x + threadIdx.x;
    if (e >= NE) return;
    int r = (int)edge_type[e];
    int dst = (int)edge_index[NE + e];
    atomicAdd(&counts[r * NN + dst], 1u);
    atomicAdd(&meta[r], 1u);
}

__global__ void rel_scan_kernel(unsigned int* __restrict__ meta) {
    if (threadIdx.x == 0 && blockIdx.x == 0) {
        unsigned int acc = 0;
        for (int r = 0; r < N_REL; ++r) {
            meta[16 + r] = acc;
            acc += meta[r];
            meta[8 + r] = 0u; // reset cursors
        }
        meta[16 + N_REL] = acc;
    }
}

__global__ void edge_bucket_kernel(const long long* __restrict__ edge_type,
                                   unsigned int* __restrict__ meta,
                                   unsigned int* __restrict__ bucket) {
    int e = blockIdx.x * blockDim.x + threadIdx.x;
    if (e >= NE) return;
    int r = (int)edge_type[e];
    unsigned int pos = atomicAdd(&meta[8 + r], 1u);
    bucket[meta[16 + r] + pos] = (unsigned int)e;
}

// ---------------- normalized scatter (aggregate-then-transform) ----------------
// agg[dst, :] += feat[src, :] / counts[r*NN + dst]  for edges of relation r.
// One thread = one float4 chunk of one edge (128 features -> 32 chunks).

__global__ void scatter_rel_kernel(const float* __restrict__ feat,
                                   const long long* __restrict__ edge_index,
                                   const unsigned int* __restrict__ counts,
                                   const unsigned int* __restrict__ meta,
                                   const unsigned int* __restrict__ bucket,
                                   float* __restrict__ agg,
                                   int r) {
    long long tid = (long long)blockIdx.x * blockDim.x + threadIdx.x;
    int el = (int)(tid >> 5);     // edge index within this relation's segment
    int chunk = (int)(tid & 31);  // which float4 of 128 features
    unsigned int base = meta[16 + r];
    unsigned int len = meta[16 + r + 1] - base;
    if (el >= (int)len) return;
    int e = (int)bucket[base + el];
    int src = (int)edge_index[e];
    int dst = (int)edge_index[NE + e];
    float scale = 1.0f / (float)counts[r * NN + dst];
    const float4 v = ((const float4*)(feat + (size_t)src * 128))[chunk];
    float* o = agg + (size_t)dst * 128 + chunk * 4;
    atomicAdd(o + 0, v.x * scale);
    atomicAdd(o + 1, v.y * scale);
    atomicAdd(o + 2, v.z * scale);
    atomicAdd(o + 3, v.w * scale);
}

// ---------------- fp32 WMMA GEMM: C (+)= A[MxK] * B[K x TN*16] (+ bias) ----------
// Block = 128 threads = 4 waves. B (K x TN*16) is staged once in LDS and shared
// by all waves. Each wave owns a 16-row strip of A/C and computes ALL TN column
// tiles, so each A fragment load feeds TN back-to-back WMMAs.
// Wave-uniform guards only after the barrier -> EXEC all-ones at every WMMA.

template <int TN>
__global__ void gemm_wmma_f32_kernel(const float* __restrict__ A, int lda,
                                     const float* __restrict__ B,
                                     float* __restrict__ C,
                                     int M, int K,
                                     const float* __restrict__ bias,
                                     int accumulate) {
    extern __shared__ float Bs[];              // K x (TN*16) floats
    const int ldb = TN * 16;                   // == ldc

    // cooperative global -> LDS stage of B (float4 granularity)
    {
        int total4 = (K * ldb) >> 2;
        for (int i = threadIdx.x; i < total4; i += blockDim.x)
            ((float4*)Bs)[i] = ((const float4*)B)[i];
    }
    __syncthreads();

    int wave = blockIdx.x * (blockDim.x >> 5) + (threadIdx.x >> 5);
    if (wave >= (M >> 4)) return;              // uniform per wave

    int lane = threadIdx.x & 31;
    int rc = lane & 15;                        // A row / B,C column within tile
    int sel = lane >> 4;                       // K-pair select (A/B); M-half (C/D)

    const float* Arow = A + (size_t)(wave * 16 + rc) * lda;

    v8f c[TN];
    if (accumulate) {
#pragma unroll
        for (int t = 0; t < TN; ++t)
#pragma unroll
            for (int i = 0; i < 8; ++i)
                c[t][i] = C[(size_t)(wave * 16 + i + 8 * sel) * ldb + t * 16 + rc];
    } else {
#pragma unroll
        for (int t = 0; t < TN; ++t) {
            float bv = bias ? bias[t * 16 + rc] : 0.0f;
#pragma unroll
            for (int i = 0; i < 8; ++i) c[t][i] = bv;
        }
    }

    for (int k0 = 0; k0 < K; k0 += 4) {
        int ka = k0 + 2 * sel;
        v2f a;
        a[0] = Arow[ka];
        a[1] = Arow[ka + 1];
#pragma unroll
        for (int t = 0; t < TN; ++t) {
            v2f b;
            b[0] = Bs[ka * ldb + t * 16 + rc];
            b[1] = Bs[(ka + 1) * ldb + t * 16 + rc];
            c[t] = __builtin_amdgcn_wmma_f32_16x16x4_f32(
                /*neg_a=*/false, a, /*neg_b=*/false, b,
                /*c_mod=*/(short)0, c[t], /*reuse_a=*/false, /*reuse_b=*/false);
        }
    }

#pragma unroll
    for (int t = 0; t < TN; ++t)
#pragma unroll
        for (int i = 0; i < 8; ++i)
            C[(size_t)(wave * 16 + i + 8 * sel) * ldb + t * 16 + rc] = c[t][i];
}

// ---------------- launch ----------------

extern "C" void kernel_launch(void* const* d_in, const int* in_sizes, int n_in,
                              void* d_out, int out_size, void* d_ws, size_t ws_size,
                              hipStream_t stream) {
    const float* x         = (const float*)d_in[0];
    const long long* eidx  = (const long long*)d_in[1]; // [2, NE] int64
    const long long* etype = (const long long*)d_in[2]; // [NE] int64
    const float* W1        = (const float*)d_in[3];     // [8,128,128]
    const float* root1     = (const float*)d_in[4];     // [128,128]
    const float* b1        = (const float*)d_in[5];     // [128]
    const float* W2        = (const float*)d_in[6];     // [8,128,64]
    const float* root2     = (const float*)d_in[7];     // [128,64]
    const float* b2        = (const float*)d_in[8];     // [64]
    float* out             = (float*)d_out;             // [NN, 64]

    // workspace layout (bytes): counts | meta | bucket | agg | h  (~56 MB)
    char* ws = (char*)d_ws;
    unsigned int* counts = (unsigned int*)ws;             // 8*NN u32 = 1,600,000 B
    unsigned int* meta   = (unsigned int*)(ws + 1600000); // 64 u32
    unsigned int* bucket = (unsigned int*)(ws + 1600256); // NE u32 = 3,200,000 B
    float* agg           = (float*)(ws + 4800256);        // NN*128 f32 = 25,600,000 B
    float* h             = (float*)(ws + 30400256);       // NN*128 f32

    const int T = 256;
    const int MT = NN / 16;                  // 3125 row tiles
    const int GB = (MT + 3) / 4;             // 4 waves (row strips) per block
    const int SH8 = 128 * 128 * 4;           // LDS bytes for B, TN=8 (64 KB)
    const int SH4 = 128 * 64 * 4;            // LDS bytes for B, TN=4 (32 KB)

    // edge preprocessing (counts + relation buckets), reused by both layers
    zero_u32_kernel<<<(8 * NN + 64 + T - 1) / T, T, 0, stream>>>(counts, 8 * NN + 64);
    edge_hist_kernel<<<(NE + T - 1) / T, T, 0, stream>>>(eidx, etype, counts, meta);
    rel_scan_kernel<<<1, 32, 0, stream>>>(meta);
    edge_bucket_kernel<<<(NE + T - 1) / T, T, 0, stream>>>(etype, meta, bucket);

    long long sc_threads = (long long)NE * 32;
    int sc_blocks = (int)((sc_threads + T - 1) / T);

    // ---- layer 1: h = relu( sum_r mean_r(x) @ W1[r] + x @ root1 + b1 ) ----
    gemm_wmma_f32_kernel<8><<<GB, 128, SH8, stream>>>(
        x, IN_DIM, root1, h, NN, IN_DIM, b1, 0);
    for (int r = 0; r < N_REL; ++r) {
        zero_f32_kernel<<<(NN * 128 + T - 1) / T, T, 0, stream>>>(agg, NN * 128);
        scatter_rel_kernel<<<sc_blocks, T, 0, stream>>>(x, eidx, counts, meta, bucket, agg, r);
        gemm_wmma_f32_kernel<8><<<GB, 128, SH8, stream>>>(
            agg, HID_DIM, W1 + (size_t)r * IN_DIM * HID_DIM, h, NN, IN_DIM, nullptr, 1);
    }
    relu_f32_kernel<<<(NN * HID_DIM + T - 1) / T, T, 0, stream>>>(h, NN * HID_DIM);

    // ---- layer 2: out = sum_r mean_r(h) @ W2[r] + h @ root2 + b2 ----
    gemm_wmma_f32_kernel<4><<<GB, 128, SH4, stream>>>(
        h, HID_DIM, root2, out, NN, HID_DIM, b2, 0);
    for (int r = 0; r < N_REL; ++r) {
        zero_f32_kernel<<<(NN * 128 + T - 1) / T, T, 0, stream>>>(agg, NN * 128);
        scatter_rel_kernel<<<sc_blocks, T, 0, stream>>>(h, eidx, counts, meta, bucket, agg, r);
        gemm_wmma_f32_kernel<4><<<GB, 128, SH4, stream>>>(
            agg, HID_DIM, W2 + (size_t)r * HID_DIM * OUT_DIM, out, NN, HID_DIM, nullptr, 1);
    }
}